// SwAVLoss_8839042695803
// MI455X (gfx1250) — compile-verified
//
#include <hip/hip_runtime.h>
#include <math.h>

// SwAV loss on gfx1250 (MI455X).
// p1: [8*1024, 3000] fp32. Output: scalar fp32 loss.
// Sinkhorn factorized into scaling vectors u (K=3000) and v (B=1024);
// row-pass GEMV accumulated with V_WMMA_F32_16X16X4_F32.
// Native exp/log (v_exp_f32/v_log_f32): args are bounded (|x| < ~8), so the
// precise-libm range fixups are dead weight in the exp-bound inner loops.

#define BS 1024
#define KP 3000
#define NCROPS 8
#define EPS_INV 20.0f   // 1/0.05
#define T_INV   10.0f   // 1/0.1

typedef float v2f __attribute__((ext_vector_type(2)));
typedef float v8f __attribute__((ext_vector_type(8)));

__device__ __forceinline__ float fast_exp(float x) { return __expf(x); }  // v_mul + v_exp_f32
__device__ __forceinline__ float fast_log(float x) { return __logf(x); }  // v_log_f32 + v_mul

__device__ __forceinline__ float block_reduce_sum_256(float v, float* sm) {
  int t = threadIdx.x;
  sm[t] = v; __syncthreads();
  for (int s = 128; s > 0; s >>= 1) { if (t < s) sm[t] += sm[t + s]; __syncthreads(); }
  float r = sm[0]; __syncthreads();
  return r;
}

// ---- per-row logsumexp of p1[r,:] / T  (all 8192 rows) ----
__global__ void rowstats_kernel(const float* __restrict__ p1, float* __restrict__ rowoff) {
  __shared__ float sm[256];
  int r = blockIdx.x;
  int t = threadIdx.x;
  const float* row = p1 + (size_t)r * KP;
  float mx = -3.4e38f;
  for (int k = t; k < KP; k += 256) mx = fmaxf(mx, row[k] * T_INV);
  sm[t] = mx; __syncthreads();
  for (int s = 128; s > 0; s >>= 1) { if (t < s) sm[t] = fmaxf(sm[t], sm[t + s]); __syncthreads(); }
  float m = sm[0]; __syncthreads();
  float acc = 0.f;
  for (int k = t; k < KP; k += 256) acc += fast_exp(row[k] * T_INV - m);
  float ssum = block_reduce_sum_256(acc, sm);
  if (t == 0) rowoff[r] = m + fast_log(ssum);
}

// ---- Sinkhorn row pass via WMMA: rs_k = sum_b exp(pc[b,k]/eps) * v_b ----
// One wave per 16-prototype tile. A[m,k4] = E[k0+m, b0+k4] (coalesced k loads),
// B[k4,n] = v[b0+k4] broadcast; D accumulates over 256 chained WMMAs.
// first=1: v==1, write raw rs.  first=0: write u_k = 1/(K*rs_k) directly.
__global__ void sinkhorn_rowpass_wmma(const float* __restrict__ pc,
                                      const float* __restrict__ vvec,
                                      float* __restrict__ outp, int first) {
  __shared__ float vsh[BS];
  int lane = threadIdx.x;                 // wave32, block == 1 wave
  for (int i = lane; i < BS; i += 32) vsh[i] = first ? 1.0f : vvec[i];
  __syncthreads();

  int m   = lane & 15;
  int hi  = lane >> 4;                    // 0: K-slots 0,1   1: K-slots 2,3
  int k0  = blockIdx.x * 16;
  int kidx = k0 + m;
  bool valid = kidx < KP;
  int kc = valid ? kidx : 0;

  v8f d = {};
  for (int b0 = 0; b0 < BS; b0 += 4) {
    int rA = b0 + 2 * hi;
    float e0 = fast_exp(pc[(size_t)rA       * KP + kc] * EPS_INV);
    float e1 = fast_exp(pc[(size_t)(rA + 1) * KP + kc] * EPS_INV);
    v2f a; a.x = valid ? e0 : 0.0f; a.y = valid ? e1 : 0.0f;   // A 16x4 layout (K=0,2 | K=1,3)
    v2f b; b.x = vsh[rA];           b.y = vsh[rA + 1];         // B 4x16 broadcast of v
    d = __builtin_amdgcn_wmma_f32_16x16x4_f32(false, a, false, b, (short)0, d, false, false);
  }

  // D layout: VGPR r -> (M=r, N=lane) lanes 0-15 ; (M=8+r, N=lane-16) lanes 16-31.
#pragma unroll
  for (int r = 0; r < 8; ++r) {
    float val = d[r];
    if (lane == 0) {
      int k = k0 + r;
      if (k < KP) outp[k] = first ? val : 1.0f / ((float)KP * val);
    } else if (lane == 16) {
      int k = k0 + 8 + r;
      if (k < KP) outp[k] = first ? val : 1.0f / ((float)KP * val);
    }
  }
}

// ---- fold global sum into first row step: S = sum(rs0); u_k = S/(K*rs0_k) ----
__global__ void finalize_u_first_kernel(const float* __restrict__ rs, float* __restrict__ u) {
  __shared__ float sm[1024];
  int t = threadIdx.x;
  float acc = 0.f;
  for (int k = t; k < KP; k += 1024) acc += rs[k];
  sm[t] = acc; __syncthreads();
  for (int s = 512; s > 0; s >>= 1) { if (t < s) sm[t] += sm[t + s]; __syncthreads(); }
  float S = sm[0]; __syncthreads();
  for (int k = t; k < KP; k += 1024) u[k] = S / ((float)KP * rs[k]);
}

// ---- Sinkhorn col pass: v_b = 1 / (B * sum_k exp(pc[b,k]/eps) * u_k) ----
__global__ void sinkhorn_colpass(const float* __restrict__ pc, const float* __restrict__ u,
                                 float* __restrict__ vvec) {
  __shared__ float sm[256];
  int b = blockIdx.x;
  int t = threadIdx.x;
  const float* row = pc + (size_t)b * KP;
  float acc = 0.f;
  for (int k = t; k < KP; k += 256) acc += fast_exp(row[k] * EPS_INV) * u[k];
  float cs = block_reduce_sum_256(acc, sm);
  if (t == 0) vvec[b] = 1.0f / ((float)BS * cs);
}

// ---- per-batch loss contribution for one crop ----
// contrib[b] = (1/T)*B*v_b * sum_k E_kb*u_k*(sum_{v!=crop} p1_v[b,k]) - sum_{v!=crop} rowoff
__global__ void final_dot_kernel(const float* __restrict__ p1, int crop,
                                 const float* __restrict__ u, const float* __restrict__ vvec,
                                 const float* __restrict__ rowoff, float* __restrict__ contrib) {
  __shared__ float sm[256];
  int b = blockIdx.x;
  int t = threadIdx.x;
  const float* rowc = p1 + ((size_t)crop * BS + b) * KP;
  float acc = 0.f;
  for (int k = t; k < KP; k += 256) {
    float pcv = rowc[k];
    float s8 = 0.f;
#pragma unroll
    for (int v = 0; v < NCROPS; ++v) s8 += p1[((size_t)v * BS + b) * KP + k];
    acc += fast_exp(pcv * EPS_INV) * u[k] * (s8 - pcv);
  }
  float tb = block_reduce_sum_256(acc, sm);
  if (t == 0) {
    float d = (float)BS * vvec[b] * tb * T_INV;
    float ro = 0.f;
    for (int v = 0; v < NCROPS; ++v) ro += rowoff[(size_t)v * BS + b];
    ro -= rowoff[(size_t)crop * BS + b];
    contrib[b] = d - ro;
  }
}

// ---- loss = -sum(contrib) / (1024 * 7 * 2) ----
__global__ void final_reduce_kernel(const float* __restrict__ contrib, float* __restrict__ out) {
  __shared__ float sm[1024];
  int t = threadIdx.x;
  sm[t] = contrib[t] + contrib[t + BS]; __syncthreads();
  for (int s = 512; s > 0; s >>= 1) { if (t < s) sm[t] += sm[t + s]; __syncthreads(); }
  if (t == 0) out[0] = -sm[0] / ((float)BS * (float)(NCROPS - 1) * 2.0f);
}

extern "C" void kernel_launch(void* const* d_in, const int* in_sizes, int n_in,
                              void* d_out, int out_size, void* d_ws, size_t ws_size,
                              hipStream_t stream) {
  (void)in_sizes; (void)n_in; (void)out_size; (void)ws_size;
  const float* p1 = (const float*)d_in[2];   // inputs: z1, z2, p1, p2

  float* ws      = (float*)d_ws;             // ~70 KB total
  float* rowoff  = ws;                        // 8192
  float* rs      = ws + 8192;                 // 3008
  float* u       = ws + 11200;                // 3008
  float* vv      = ws + 14208;                // 1024
  float* contrib = ws + 15232;                // 2048

  rowstats_kernel<<<dim3(BS * NCROPS), dim3(256), 0, stream>>>(p1, rowoff);

  const int KTILES = (KP + 15) / 16;          // 188
  for (int crop = 0; crop < 2; ++crop) {
    const float* pc = p1 + (size_t)crop * BS * KP;
    // iter 1 row (with global-sum fold), then col
    sinkhorn_rowpass_wmma<<<dim3(KTILES), dim3(32), 0, stream>>>(pc, vv, rs, 1);
    finalize_u_first_kernel<<<dim3(1), dim3(1024), 0, stream>>>(rs, u);
    sinkhorn_colpass<<<dim3(BS), dim3(256), 0, stream>>>(pc, u, vv);
    // iters 2,3
    for (int it = 0; it < 2; ++it) {
      sinkhorn_rowpass_wmma<<<dim3(KTILES), dim3(32), 0, stream>>>(pc, vv, u, 0);
      sinkhorn_colpass<<<dim3(BS), dim3(256), 0, stream>>>(pc, u, vv);
    }
    final_dot_kernel<<<dim3(BS), dim3(256), 0, stream>>>(p1, crop, u, vv, rowoff, contrib + crop * BS);
  }
  final_reduce_kernel<<<dim3(1), dim3(1024), 0, stream>>>(contrib, (float*)d_out);
}